// HyperNet_39041252721062
// MI455X (gfx1250) — compile-verified
//
#include <hip/hip_runtime.h>

typedef __attribute__((ext_vector_type(2))) float v2f;
typedef __attribute__((ext_vector_type(8))) float v8f;

#define BATCH   64
#define DFEAT   512
#define EMB     16
#define KCODES  512
#define P_TOTAL 1482625
#define LDSPAD  18   // padded row stride (floats): conflict-free, keeps 8B alignment

// ---------------------------------------------------------------------------
// Kernel 1: embed = feat @ fc_w^T + fc_b  (WMMA f32 16x16x4, 4 waves),
// then VQ argmin + gather + loss. quantized -> d_ws, loss -> loss_out.
// ---------------------------------------------------------------------------
__global__ __launch_bounds__(128) void encode_vq_kernel(
    const float* __restrict__ feat, const float* __restrict__ fc_w,
    const float* __restrict__ fc_b, const float* __restrict__ codebook,
    float* __restrict__ qout, float* __restrict__ loss_out)
{
    __shared__ float emb[BATCH * LDSPAD];
    __shared__ float row_err[BATCH];

    const int tid  = threadIdx.x;
    const int lane = tid & 31;
    const int wave = tid >> 5;        // M-tile: rows 16*wave .. 16*wave+15
    const int n    = lane & 15;       // A: M index / B: N index
    const int half = lane >> 4;       // A/B: K-half ; D: M-half

    // A: feat row (16*wave + n), B: fc_w^T column n == fc_w row n
    const float* arow = feat + (size_t)(16 * wave + n) * DFEAT;
    const float* brow = fc_w + (size_t)n * DFEAT;

    v8f acc = {};
    for (int kk = 0; kk < DFEAT / 4; ++kk) {
        const int kcol = 4 * kk + 2 * half;              // even -> 8B aligned
        const v2f a = *reinterpret_cast<const v2f*>(arow + kcol);
        const v2f b = *reinterpret_cast<const v2f*>(brow + kcol);
        acc = __builtin_amdgcn_wmma_f32_16x16x4_f32(
            false, a, false, b, (short)0, acc, false, false);
    }

    const float bias = fc_b[n];
    for (int r = 0; r < 8; ++r) {
        const int brows = 16 * wave + r + 8 * half;      // batch row of D[r]
        emb[brows * LDSPAD + n] = acc[r] + bias;
    }
    __syncthreads();

    // VQ: one thread per batch row (first 2 waves active)
    if (tid < BATCH) {
        float e[EMB];
        for (int j = 0; j < EMB; ++j) e[j] = emb[tid * LDSPAD + j];
        int   best  = 0;
        float bestd = 3.402823466e38f;
        for (int c = 0; c < KCODES; ++c) {
            const float* cb = codebook + c * EMB;
            float d = 0.f;
            for (int j = 0; j < EMB; ++j) {
                const float t = e[j] - cb[j];
                d = fmaf(t, t, d);
            }
            if (d < bestd) { bestd = d; best = c; }   // strict < == first-min (argmin)
        }
        const float* cb = codebook + best * EMB;
        float err = 0.f;
        for (int j = 0; j < EMB; ++j) {
            const float qv = cb[j];
            qout[tid * EMB + j] = qv;                 // quantized_st == quantized fwd
            const float t = qv - e[j];
            err = fmaf(t, t, err);
        }
        row_err[tid] = err;
    }
    __syncthreads();

    if (tid == 0) {
        float s = 0.f;
        for (int i = 0; i < BATCH; ++i) s += row_err[i];
        // loss = q_latent + 0.25*e_latent ; both equal mse(quantized, embed) fwd
        *loss_out = 1.25f * s / (float)(BATCH * EMB);
    }
}

// ---------------------------------------------------------------------------
// Kernel 2: gen_weights = Q[64,16] @ Wgen[16,P].  Memory-bound (reads 95MB,
// writes 380MB -> ~20us floor @ 23.3TB/s). One wave per 16-column tile,
// 4 M-tiles x 4 K-steps = 16 V_WMMA_F32_16X16X4_F32 per wave.
// Wgen loads and output stores are one-touch (475MB > 192MB L2) -> TH=NT.
// ---------------------------------------------------------------------------
__global__ __launch_bounds__(256) void hypergen_kernel(
    const float* __restrict__ Wgen, const float* __restrict__ qws,
    float* __restrict__ out)
{
    __shared__ float q[BATCH * LDSPAD];

    const int tid = threadIdx.x;
    // Stage Q (64x16) into padded LDS; 1024 elems / 256 threads, no divergence.
    // Q is reused by every block -> default (RT) policy keeps it L2-resident.
    for (int i = tid; i < BATCH * EMB; i += 256)
        q[(i >> 4) * LDSPAD + (i & 15)] = qws[i];
    __syncthreads();

    const int lane = tid & 31;
    const int wave = tid >> 5;
    const int tile = blockIdx.x * 8 + wave;           // wave-uniform
    const int NT   = (P_TOTAL + 15) / 16;             // 92665 tiles
    if (tile >= NT) return;                           // uniform -> EXEC stays full

    const int  n    = lane & 15;
    const int  half = lane >> 4;
    const int  p    = tile * 16 + n;
    const long pc   = (p < P_TOTAL) ? p : (P_TOTAL - 1);  // clamp tail loads

    // A fragments from LDS: lane holds Q[16t + n][4kk + 2*half .. +1]
    v2f afrag[4][4];
    for (int t = 0; t < 4; ++t)
        for (int kk = 0; kk < 4; ++kk)
            afrag[t][kk] = *reinterpret_cast<const v2f*>(
                &q[(16 * t + n) * LDSPAD + 4 * kk + 2 * half]);

    v8f acc[4] = {{}, {}, {}, {}};
    for (int kk = 0; kk < 4; ++kk) {
        const int kr = 4 * kk + 2 * half;             // B: K rows 2 apart per lane-half
        v2f b;
        b[0] = __builtin_nontemporal_load(&Wgen[(long)kr * P_TOTAL + pc]);
        b[1] = __builtin_nontemporal_load(&Wgen[(long)(kr + 1) * P_TOTAL + pc]);
        for (int t = 0; t < 4; ++t)
            acc[t] = __builtin_amdgcn_wmma_f32_16x16x4_f32(
                false, afrag[t][kk], false, b, (short)0, acc[t], false, false);
    }

    if (p < P_TOTAL) {                                 // predicated stores (non-WMMA)
        for (int t = 0; t < 4; ++t)
            for (int r = 0; r < 8; ++r) {
                const long row = 16 * t + r + 8 * half;
                __builtin_nontemporal_store(acc[t][r], &out[row * (long)P_TOTAL + p]);
            }
    }
}

// ---------------------------------------------------------------------------
extern "C" void kernel_launch(void* const* d_in, const int* in_sizes, int n_in,
                              void* d_out, int out_size, void* d_ws, size_t ws_size,
                              hipStream_t stream) {
    const float* feat     = (const float*)d_in[0];
    const float* fc_w     = (const float*)d_in[1];
    const float* fc_b     = (const float*)d_in[2];
    const float* codebook = (const float*)d_in[3];
    const float* Wgen     = (const float*)d_in[4];

    float* out      = (float*)d_out;
    float* loss_out = out + (out_size - 1);           // loss is last element
    float* qws      = (float*)d_ws;                   // quantized [64,16]

    encode_vq_kernel<<<1, 128, 0, stream>>>(feat, fc_w, fc_b, codebook, qws, loss_out);

    const int NT = (P_TOTAL + 15) / 16;               // 92665 column tiles
    hypergen_kernel<<<(NT + 7) / 8, 256, 0, stream>>>(Wgen, qws, out);
}